// StructuralVPhi_90254442758428
// MI455X (gfx1250) — compile-verified
//
#include <hip/hip_runtime.h>
#include <hip/hip_bf16.h>

// Problem constants (match reference)
#define NB 8
#define NT 512
#define ND 128
#define NDL 16
#define NK 8
#define NHPHI 32
#define NHTH 32
#define EPS2F 0.0001f   // 0.01^2

// LDS row strides (dwords): +4 pad to de-conflict 64-bank LDS
#define HSTR 132   // 128 + 4
#define PSTR 36    // 32 + 4
#define LSTR 20    // 16 + 4

typedef float v2f __attribute__((ext_vector_type(2)));
typedef float v8f __attribute__((ext_vector_type(8)));
typedef unsigned int u32x4 __attribute__((ext_vector_type(4)));
typedef int i32x4 __attribute__((ext_vector_type(4)));
typedef int i32x8 __attribute__((ext_vector_type(8)));

#if defined(__has_builtin)
#if __has_builtin(__builtin_amdgcn_tensor_load_to_lds) && __has_builtin(__builtin_amdgcn_s_wait_tensorcnt)
#define USE_TDM 1
#endif
#endif
#ifndef USE_TDM
#define USE_TDM 0
#endif

__device__ __forceinline__ float gelu_exact(float x) {
    // jax.nn.gelu(approximate=False): 0.5*x*(1+erf(x/sqrt(2)))
    return 0.5f * x * (1.0f + erff(x * 0.70710678118654752440f));
}

__device__ __forceinline__ float softplus_f(float x) {
    return (x > 20.0f) ? x : log1pf(expf(x));
}

#if USE_TDM
// 1-D contiguous TDM copy of n f32 elements global->LDS, with optional LDS
// row padding inserted by the DMA engine (pad `pad_amt_code+1` dwords after
// every 2^(pad_code+1) dwords stored).  D# packing per CDNA5 ISA ch.8.
__device__ __forceinline__ void tdm_load_1d(unsigned lds_off, const void* gsrc,
                                            unsigned n, bool pad_en,
                                            unsigned pad_code, unsigned pad_amt_code) {
    const unsigned long long ga = (unsigned long long)(uintptr_t)gsrc;
    u32x4 g0 = {
        1u,                                                  // count=1, user D#
        lds_off,                                             // lds_addr (bytes)
        (unsigned)(ga & 0xffffffffull),                      // global_addr lo
        (unsigned)((ga >> 32) & 0x01ffffffull) | (2u << 30)  // global_addr hi | type=2
    };
    const unsigned d0 = (2u << 16) |                         // data_size = 4B, wg_mask=0
        (pad_en ? ((1u << 20) | (pad_code << 22) | (pad_amt_code << 25)) : 0u);
    i32x8 g1 = {
        (int)d0,
        (int)((n & 0xffffu) << 16),          // tensor_dim0[15:0]
        (int)((n >> 16) | (1u << 16)),       // tensor_dim0[31:16] | tensor_dim1=1
        (int)((n & 0xffffu) << 16),          // tile_dim0 = n
        (int)1,                              // tile_dim1 = 1, tile_dim2 = 0
        (int)n,                              // tensor_dim0_stride lo
        0, 0
    };
    i32x4 gz = {0, 0, 0, 0};
#if __clang_major__ >= 23
    i32x8 gz8 = {0, 0, 0, 0, 0, 0, 0, 0};
    __builtin_amdgcn_tensor_load_to_lds(g0, g1, gz, gz, gz8, 0);
#else
    __builtin_amdgcn_tensor_load_to_lds(g0, g1, gz, gz, 0);
#endif
}
#endif

// ---------------------------------------------------------------------------
// Kernel 1: per-row projections + squared norms.
// ---------------------------------------------------------------------------
__global__ __launch_bounds__(256) void svphi_proj_kernel(
    const float* __restrict__ h, const float* __restrict__ hsrc,
    const float* __restrict__ W_l, const float* __restrict__ W_th,
    const float* __restrict__ wq, const float* __restrict__ wsm,
    const float* __restrict__ wd, const float* __restrict__ b1,
    float* __restrict__ lq, float* __restrict__ ls,
    float* __restrict__ pt, float* __restrict__ pu,
    float* __restrict__ nlq, float* __restrict__ nls,
    float* __restrict__ nh, float* __restrict__ nhs)
{
    const int row = blockIdx.x * blockDim.x + threadIdx.x;   // 0 .. B*T-1
    const float* hr = h    + (size_t)row * ND;
    const float* sr = hsrc + (size_t)row * ND;

    float lqv[NDL]; float lsv[NDL];
    float tq[NK];   float ts[NK];
    #pragma unroll
    for (int j = 0; j < NDL; ++j) { lqv[j] = 0.0f; lsv[j] = 0.0f; }
    #pragma unroll
    for (int j = 0; j < NK; ++j)  { tq[j] = 0.0f;  ts[j] = 0.0f;  }
    float nhv = 0.0f, nsv = 0.0f;

    for (int d = 0; d < ND; ++d) {
        const float a = hr[d];
        const float c = sr[d];
        nhv = fmaf(a, a, nhv);
        nsv = fmaf(c, c, nsv);
        const float* wl = W_l  + d * NDL;
        #pragma unroll
        for (int j = 0; j < NDL; ++j) {
            lqv[j] = fmaf(a, wl[j], lqv[j]);
            lsv[j] = fmaf(c, wl[j], lsv[j]);
        }
        const float* wt = W_th + d * NK;
        #pragma unroll
        for (int j = 0; j < NK; ++j) {
            tq[j] = fmaf(a, wt[j], tq[j]);
            ts[j] = fmaf(c, wt[j], ts[j]);
        }
    }

    float nq = 0.0f, ns2 = 0.0f;
    #pragma unroll
    for (int j = 0; j < NDL; ++j) {
        nq  = fmaf(lqv[j], lqv[j], nq);
        ns2 = fmaf(lsv[j], lsv[j], ns2);
        lq[(size_t)row * NDL + j] = lqv[j];
        ls[(size_t)row * NDL + j] = lsv[j];
    }

    #pragma unroll 4
    for (int j = 0; j < NHTH; ++j) {
        float aq = b1[j];
        float au = 0.0f;
        #pragma unroll
        for (int k = 0; k < NK; ++k) {
            const float wqk = wq[k * NHTH + j];
            const float wsk = wsm[k * NHTH + j];
            const float wdk = wd[k * NHTH + j];
            aq = fmaf(tq[k], wqk + wdk, aq);
            au = fmaf(ts[k], wsk - wdk, au);
        }
        pt[(size_t)row * NHTH + j] = aq;
        pu[(size_t)row * NHTH + j] = au;
    }

    nlq[row] = nq;
    nls[row] = ns2;
    nh[row]  = nhv;
    nhs[row] = nsv;
}

// ---------------------------------------------------------------------------
// Kernel 2: TDM -> LDS -> ds_load -> V_WMMA_F32_16X16X4_F32 pipeline.
// Block = 256 threads = 8 waves; macro tile 32(t) x 64(s); 16x16 tile/wave.
// ---------------------------------------------------------------------------
__global__ __launch_bounds__(256) void svphi_pair_kernel(
    const float* __restrict__ h, const float* __restrict__ hsrc,
    const float* __restrict__ lq, const float* __restrict__ ls,
    const float* __restrict__ pt, const float* __restrict__ pu,
    const float* __restrict__ nlq, const float* __restrict__ nls,
    const float* __restrict__ nh, const float* __restrict__ nhs,
    const float* __restrict__ phi1w, const float* __restrict__ phi1b,
    const float* __restrict__ phi2w, const float* __restrict__ phi2bp,
    const float* __restrict__ w2w, const float* __restrict__ w2bp,
    float* __restrict__ out)
{
    __shared__ __align__(16) float s_ht[32 * HSTR];   // h rows (t side)
    __shared__ __align__(16) float s_hs[64 * HSTR];   // h_src rows (s side)
    __shared__ __align__(16) float s_lq[32 * LSTR];   // l_q rows
    __shared__ __align__(16) float s_ls[64 * LSTR];   // l_s rows
    __shared__ __align__(16) float s_pt[32 * PSTR];   // proj_t rows
    __shared__ __align__(16) float s_pu[64 * PSTR];   // proj_u rows
    __shared__ __align__(16) float s_nlq[32];
    __shared__ __align__(16) float s_nh[32];
    __shared__ __align__(16) float s_nls[64];
    __shared__ __align__(16) float s_nhs[64];

    const int tid  = threadIdx.x;
    const int lane = tid & 31;
    const int wave = tid >> 5;
    const int l2   = lane & 15;
    const int hi   = lane >> 4;
    const int b    = blockIdx.z;
    const int t0   = blockIdx.y * 32;     // block's t range  [t0, t0+32)
    const int s0   = blockIdx.x * 64;     // block's s range  [s0, s0+64)
    const size_t bT = (size_t)b * NT;

#if USE_TDM
    if (wave == 0) {
        // One descriptor per staged region; DMA engine inserts the +4 dword
        // row padding (codes: interval 2^(c+1) dwords, amount a+1 dwords).
        tdm_load_1d((unsigned)(uintptr_t)s_ht,  h    + (bT + t0) * ND, 32 * ND,  true, 6, 3);
        tdm_load_1d((unsigned)(uintptr_t)s_hs,  hsrc + (bT + s0) * ND, 64 * ND,  true, 6, 3);
        tdm_load_1d((unsigned)(uintptr_t)s_lq,  lq   + (bT + t0) * NDL, 32 * NDL, true, 3, 3);
        tdm_load_1d((unsigned)(uintptr_t)s_ls,  ls   + (bT + s0) * NDL, 64 * NDL, true, 3, 3);
        tdm_load_1d((unsigned)(uintptr_t)s_pt,  pt   + (bT + t0) * NHTH, 32 * NHTH, true, 4, 3);
        tdm_load_1d((unsigned)(uintptr_t)s_pu,  pu   + (bT + s0) * NHTH, 64 * NHTH, true, 4, 3);
        tdm_load_1d((unsigned)(uintptr_t)s_nlq, nlq + bT + t0, 32, false, 0, 0);
        tdm_load_1d((unsigned)(uintptr_t)s_nh,  nh  + bT + t0, 32, false, 0, 0);
        tdm_load_1d((unsigned)(uintptr_t)s_nls, nls + bT + s0, 64, false, 0, 0);
        tdm_load_1d((unsigned)(uintptr_t)s_nhs, nhs + bT + s0, 64, false, 0, 0);
        __builtin_amdgcn_s_wait_tensorcnt(0);
    }
#else
    {
        const float* gh = h    + (bT + t0) * ND;
        const float* gs = hsrc + (bT + s0) * ND;
        for (int i = tid; i < 32 * ND; i += 256) s_ht[(i >> 7) * HSTR + (i & 127)] = gh[i];
        for (int i = tid; i < 64 * ND; i += 256) s_hs[(i >> 7) * HSTR + (i & 127)] = gs[i];
        const float* glq = lq + (bT + t0) * NDL;
        const float* gls = ls + (bT + s0) * NDL;
        for (int i = tid; i < 32 * NDL; i += 256) s_lq[(i >> 4) * LSTR + (i & 15)] = glq[i];
        for (int i = tid; i < 64 * NDL; i += 256) s_ls[(i >> 4) * LSTR + (i & 15)] = gls[i];
        const float* gpt = pt + (bT + t0) * NHTH;
        const float* gpu = pu + (bT + s0) * NHTH;
        for (int i = tid; i < 32 * NHTH; i += 256) s_pt[(i >> 5) * PSTR + (i & 31)] = gpt[i];
        for (int i = tid; i < 64 * NHTH; i += 256) s_pu[(i >> 5) * PSTR + (i & 31)] = gpu[i];
        if (tid < 32) { s_nlq[tid] = nlq[bT + t0 + tid]; s_nh[tid] = nh[bT + t0 + tid]; }
        else if (tid < 96) { int i = tid - 32; s_nls[i] = nls[bT + s0 + i]; s_nhs[i] = nhs[bT + s0 + i]; }
    }
#endif
    __syncthreads();

    const int tw = (wave >> 2) * 16;      // wave's local t base (0 or 16)
    const int sw = (wave & 3) * 16;       // wave's local s base (0,16,32,48)

    // --- WMMA over D=128 from LDS (ds_load_b64 fragments) ---
    const float* Ar = s_ht + (tw + l2) * HSTR + 2 * hi;
    const float* Br = s_hs + (sw + l2) * HSTR + 2 * hi;
    v8f acc_h = {0.0f, 0.0f, 0.0f, 0.0f, 0.0f, 0.0f, 0.0f, 0.0f};
    #pragma unroll
    for (int k = 0; k < ND; k += 4) {
        v2f av = *(const v2f*)(Ar + k);
        v2f bv = *(const v2f*)(Br + k);
        acc_h = __builtin_amdgcn_wmma_f32_16x16x4_f32(
            false, av, false, bv, (short)0, acc_h, false, false);
    }

    // --- WMMA over DL=16 from LDS ---
    const float* Al = s_lq + (tw + l2) * LSTR + 2 * hi;
    const float* Bl = s_ls + (sw + l2) * LSTR + 2 * hi;
    v8f acc_l = {0.0f, 0.0f, 0.0f, 0.0f, 0.0f, 0.0f, 0.0f, 0.0f};
    #pragma unroll
    for (int k = 0; k < NDL; k += 4) {
        v2f av = *(const v2f*)(Al + k);
        v2f bv = *(const v2f*)(Bl + k);
        acc_l = __builtin_amdgcn_wmma_f32_16x16x4_f32(
            false, av, false, bv, (short)0, acc_l, false, false);
    }

    // --- Epilogue ---
    const int su = sw + l2;                // local s in [0,64)   (N = lane&15)
    const float nls_v = s_nls[su];
    const float nhs_v = s_nhs[su];

    // proj_u row fixed per lane: hoist to registers (ds_load_b128 x8)
    float pur[NHTH];
    const float4* pu4 = (const float4*)(s_pu + su * PSTR);
    #pragma unroll
    for (int q = 0; q < NHTH / 4; ++q) {
        float4 v = pu4[q];
        pur[4 * q + 0] = v.x; pur[4 * q + 1] = v.y;
        pur[4 * q + 2] = v.z; pur[4 * q + 3] = v.w;
    }

    float d2l[8], rinv[8], accC[8], accT[8];
    #pragma unroll
    for (int r = 0; r < 8; ++r) {
        const int tl = tw + r + 8 * hi;    // local t   (M = r + 8*hi)
        d2l[r] = fmaxf(s_nlq[tl] + nls_v - 2.0f * acc_l[r], 0.0f);
        const float d2h = fmaxf(s_nh[tl] + nhs_v - 2.0f * acc_h[r], 0.0f);
        rinv[r] = 1.0f / sqrtf(d2h + EPS2F);
        accC[r] = 0.0f;
        accT[r] = 0.0f;
    }

    // q outer so each weight quad is loaded once per tile
    const float4* p1w4 = (const float4*)phi1w;
    const float4* p1b4 = (const float4*)phi1b;
    const float4* p2w4 = (const float4*)phi2w;
    const float4* w2w4 = (const float4*)w2w;
    for (int q = 0; q < NHPHI / 4; ++q) {
        const float4 w1 = p1w4[q];
        const float4 bb = p1b4[q];
        const float4 pw = p2w4[q];
        const float4 tw2 = w2w4[q];
        #pragma unroll
        for (int r = 0; r < 8; ++r) {
            const int tl = tw + r + 8 * hi;
            const float4 ptq = *(const float4*)(s_pt + tl * PSTR + 4 * q);
            const float d2 = d2l[r];
            float aC = accC[r], aT = accT[r];
            aC = fmaf(gelu_exact(fmaf(d2, w1.x, bb.x)), pw.x, aC);
            aC = fmaf(gelu_exact(fmaf(d2, w1.y, bb.y)), pw.y, aC);
            aC = fmaf(gelu_exact(fmaf(d2, w1.z, bb.z)), pw.z, aC);
            aC = fmaf(gelu_exact(fmaf(d2, w1.w, bb.w)), pw.w, aC);
            aT = fmaf(gelu_exact(ptq.x + pur[4 * q + 0]), tw2.x, aT);
            aT = fmaf(gelu_exact(ptq.y + pur[4 * q + 1]), tw2.y, aT);
            aT = fmaf(gelu_exact(ptq.z + pur[4 * q + 2]), tw2.z, aT);
            aT = fmaf(gelu_exact(ptq.w + pur[4 * q + 3]), tw2.w, aT);
            accC[r] = aC;
            accT[r] = aT;
        }
    }

    const float phi2b = phi2bp[0];
    const float w2b   = w2bp[0];
    const int s_idx = s0 + su;
    #pragma unroll
    for (int r = 0; r < 8; ++r) {
        const float c     = softplus_f(accC[r] + phi2b);
        const float Phi   = expf(-c * d2l[r]);
        const float Theta = tanhf(accT[r] + w2b);
        const int t_idx = t0 + tw + r + 8 * hi;
        __builtin_nontemporal_store(-(Theta * Phi) * rinv[r],
                                    &out[(bT + t_idx) * NT + s_idx]);
    }
}

// ---------------------------------------------------------------------------
extern "C" void kernel_launch(void* const* d_in, const int* in_sizes, int n_in,
                              void* d_out, int out_size, void* d_ws, size_t ws_size,
                              hipStream_t stream) {
    const float* h     = (const float*)d_in[0];
    const float* hsrc  = (const float*)d_in[1];
    const float* W_l   = (const float*)d_in[2];
    const float* W_th  = (const float*)d_in[3];
    const float* phi1w = (const float*)d_in[4];
    const float* phi1b = (const float*)d_in[5];
    const float* phi2w = (const float*)d_in[6];
    const float* phi2b = (const float*)d_in[7];
    const float* wq    = (const float*)d_in[8];
    const float* wsm   = (const float*)d_in[9];
    const float* wd    = (const float*)d_in[10];
    const float* b1    = (const float*)d_in[11];
    const float* w2w   = (const float*)d_in[12];
    const float* w2b   = (const float*)d_in[13];
    float* out = (float*)d_out;

    // Workspace layout (floats): lq, ls, pt, pu, nlq, nls, nh, nhs
    const size_t NR = (size_t)NB * NT;          // 4096 rows
    float* wsf = (float*)d_ws;
    float* lq  = wsf;
    float* ls  = lq  + NR * NDL;
    float* pt  = ls  + NR * NDL;
    float* pu  = pt  + NR * NHTH;
    float* nlq = pu  + NR * NHTH;
    float* nls = nlq + NR;
    float* nh  = nls + NR;
    float* nhs = nh  + NR;

    svphi_proj_kernel<<<dim3((unsigned)(NR / 256)), dim3(256), 0, stream>>>(
        h, hsrc, W_l, W_th, wq, wsm, wd, b1,
        lq, ls, pt, pu, nlq, nls, nh, nhs);

    dim3 grid2(NT / 64, NT / 32, NB);
    svphi_pair_kernel<<<grid2, dim3(256), 0, stream>>>(
        h, hsrc, lq, ls, pt, pu, nlq, nls, nh, nhs,
        phi1w, phi1b, phi2w, phi2b, w2w, w2b, out);
}